// Pixel_Unshuffle_50732153700396
// MI455X (gfx1250) — compile-verified
//
#include <hip/hip_runtime.h>
#include <hip/hip_bf16.h>

// Pixel-unshuffle r=2, fp32: (16,64,512,512) -> (16,256,256,256).
// Pure streaming permutation: ~2.15 GB total traffic, HBM-bound (~92us floor
// at 23.3 TB/s). No WMMA (0 FLOPs). CDNA5 async global->LDS loads (ASYNCcnt,
// 4x b128 per thread via instruction offsets), conflict-free ds_load_b128,
// register de-interleave, non-temporal coalesced b64 stores. 64 B of traffic
// per thread amortizes the per-thread index math ~4x vs the naive version so
// the kernel stays bandwidth-bound, not issue-bound.

#define N_  16
#define C_  64
#define H_  512
#define W_  512
#define HP_ (H_ / 2)   // 256
#define WP_ (W_ / 2)   // 256

typedef float v2f __attribute__((ext_vector_type(2)));
typedef float v4f __attribute__((ext_vector_type(4)));
typedef __attribute__((address_space(3))) float lds_float_t;

__global__ __launch_bounds__(256)
void pixel_unshuffle_r2_kernel(const float* __restrict__ in,
                               float* __restrict__ out) {
    // Block b -> (n, g, i4). Input rows 8*i4 .. 8*i4+7 of (n,g): 4096
    // contiguous floats (16 KB). Output: rows 4*i4..4*i4+3 of channels
    // g*4..g*4+3 (16 KB).
    __shared__ alignas(16) float tile[4096];

    const int t = threadIdx.x;            // 0..255
    const unsigned b = blockIdx.x;        // 0..65535
    const int i4 = b & 63;
    const int g  = (b >> 6) & 63;
    const int n  = b >> 12;

    // float index of input row 8*i4 start (8 rows are contiguous: 16 KB)
    const size_t inBase = (((size_t)n * C_ + g) * H_ + 8 * (size_t)i4) * W_;
    const float* src = in + inBase + 4 * (size_t)t;     // this lane's first 16B

    // LDS byte offset of this lane's first 16B slot (true AS3 offset)
    unsigned ldsByte =
        (unsigned)(unsigned long long)(lds_float_t*)&tile[0] + (unsigned)t * 16u;

    // Stage 1: 4x async copy global -> LDS. INST_OFFSET is added to both the
    // LDS and the global address, so one address pair serves all four.
    asm volatile(
        "global_load_async_to_lds_b128 %0, %1, off\n\t"
        "global_load_async_to_lds_b128 %0, %1, off offset:4096\n\t"
        "global_load_async_to_lds_b128 %0, %1, off offset:8192\n\t"
        "global_load_async_to_lds_b128 %0, %1, off offset:12288\n\t"
        "s_wait_asynccnt 0\n\t"
        :
        : "v"(ldsByte), "v"(src)
        : "memory");
    __syncthreads();

    // Stage 2: 4x conflict-free ds_load_b128, de-interleave, 8x NT b64 store.
    const int y  = t >> 7;        // input row parity (constant across k)
    const int w4 = t & 127;       // float4 index within a 512-float row
    const int cb = g * 4 + 2 * y; // output channel base for this parity

    // out float index of (n, cb, 4*i4, 2*w4); k advances one output row
    const size_t ob =
        ((((size_t)n * (C_ * 4) + cb) * HP_) + 4 * (size_t)i4) * WP_ + 2 * (size_t)w4;
    float* outE = out + ob;                          // x = 0 channel
    float* outO = out + ob + (size_t)HP_ * WP_;      // x = 1 channel

#pragma unroll
    for (int k = 0; k < 4; ++k) {
        const v4f a = *reinterpret_cast<const v4f*>(&tile[4 * t + 1024 * k]);
        const v2f even = {a.x, a.z};
        const v2f odd  = {a.y, a.w};
        // 32 lanes x 8B = 256B contiguous per store; zero reuse -> NT.
        __builtin_nontemporal_store(even, reinterpret_cast<v2f*>(outE + k * WP_));
        __builtin_nontemporal_store(odd,  reinterpret_cast<v2f*>(outO + k * WP_));
    }
}

extern "C" void kernel_launch(void* const* d_in, const int* in_sizes, int n_in,
                              void* d_out, int out_size, void* d_ws, size_t ws_size,
                              hipStream_t stream) {
    (void)in_sizes; (void)n_in; (void)d_ws; (void)ws_size; (void)out_size;
    const float* put = (const float*)d_in[0];
    float* out = (float*)d_out;

    const int blocks = N_ * C_ * (HP_ / 4);   // 16*64*64 = 65536
    pixel_unshuffle_r2_kernel<<<blocks, 256, 0, stream>>>(put, out);
}